// NodeFormerConv_28673201668590
// MI455X (gfx1250) — compile-verified
//
#include <hip/hip_runtime.h>
#include <climits>

typedef __attribute__((ext_vector_type(2))) float v2f;
typedef __attribute__((ext_vector_type(8))) float v8f;

#define DEVFN __device__ __forceinline__

// Problem constants (from reference setup_inputs)
constexpr int Nn   = 30000;
constexpr int CIN  = 128;
constexpr int Hh   = 4;
constexpr int Dd   = 64;
constexpr int HD   = 256;   // H*D
constexpr int Mm   = 30;    // random features
constexpr int MP   = 32;    // M padded for WMMA inner dim
constexpr int Kk   = 10;    // gumbel samples
constexpr int Ee   = 480000;
constexpr int NCHUNK = 30;  // N split for kvs reduction (30000/30 = 1000)
constexpr int NTILES = Nn / 16;  // 1875

constexpr float SCALE_QK = 0.70710678118654752f;  // tau^-0.5 * 64^-0.25 = 2 * 0.353553
constexpr float RATIO    = 0.18257418583505536f;  // 30^-0.5
constexpr float EPSV     = 1e-6f;

DEVFN v8f wmma4(v2f a, v2f b, v8f c) {
  // V_WMMA_F32_16X16X4_F32 : D = A(16x4) * B(4x16) + C(16x16), all f32
  return __builtin_amdgcn_wmma_f32_16x16x4_f32(false, a, false, b, (short)0, c, false, false);
}

// order-preserving float<->int encoding for atomicMax on floats
DEVFN int   enc_f(float f) { int i = __float_as_int(f); return i >= 0 ? i : (i ^ 0x7fffffff); }
DEVFN float dec_f(int i)   { return __int_as_float(i >= 0 ? i : (i ^ 0x7fffffff)); }

// ---------------------------------------------------------------------------
// Generic GEMM:  Y[N, Cout] = X[N, Kdim] @ W[Cout, Kdim]^T + bias
// one 16x16 output tile per wave, f32 WMMA 16x16x4 over Kdim
// ---------------------------------------------------------------------------
__global__ void gemm_xwt(const float* __restrict__ X, const float* __restrict__ W,
                         const float* __restrict__ bias, float* __restrict__ Y,
                         int Kdim, int Cout) {
  int wave = threadIdx.x >> 5;
  int rowTile = blockIdx.x * 4 + wave;
  if (rowTile >= NTILES) return;
  int rowBase = rowTile * 16;
  int colBase = blockIdx.y * 16;
  int lane = threadIdx.x & 31;
  int half = lane >> 4;       // 0: K+0/1, 1: K+2/3
  int lidx = lane & 15;

  v8f acc = {};
  for (int kk = 0; kk < Kdim; kk += 4) {
    int ka = kk + half * 2;
    const float* xp = X + (size_t)(rowBase + lidx) * Kdim + ka;
    const float* wp = W + (size_t)(colBase + lidx) * Kdim + ka;
    v2f a, b;
    a.x = xp[0]; a.y = xp[1];
    b.x = wp[0]; b.y = wp[1];
    acc = wmma4(a, b, acc);
  }
  float bv = bias[colBase + lidx];
#pragma unroll
  for (int r = 0; r < 8; ++r) {
    int m = r + half * 8;
    Y[(size_t)(rowBase + m) * Cout + colBase + lidx] = acc[r] + bv;
  }
}

// ---------------------------------------------------------------------------
// qp: FAVOR+ positive features for queries (per-(n,h) stabilizer)
// qp layout [N, H, 32] with m=30,31 zero-padded for WMMA
// ---------------------------------------------------------------------------
__global__ void qp_kernel(const float* __restrict__ Q, const float* __restrict__ proj,
                          float* __restrict__ qp) {
  __shared__ float sproj[Mm * Dd];
  for (int i = threadIdx.x; i < Mm * Dd; i += blockDim.x) sproj[i] = proj[i];
  __syncthreads();
  int unit = blockIdx.x * blockDim.x + threadIdx.x;
  if (unit >= Nn * Hh) return;
  int n = unit >> 2, h = unit & 3;
  const float* qrow = Q + (size_t)n * HD + h * Dd;
  float data[Dd];
  float diag = 0.f;
  for (int d = 0; d < Dd; ++d) { float x = qrow[d] * SCALE_QK; data[d] = x; diag += x * x; }
  diag *= 0.5f;
  float dd[Mm];
  float stab = -1e30f;
#pragma unroll 1
  for (int m = 0; m < Mm; ++m) {
    const float* pr = sproj + m * Dd;
    float s = 0.f;
    for (int d = 0; d < Dd; ++d) s += data[d] * pr[d];
    dd[m] = s; stab = fmaxf(stab, s);
  }
  float* out = qp + (size_t)unit * MP;
  for (int m = 0; m < Mm; ++m) out[m] = RATIO * (expf(dd[m] - diag - stab) + EPSV);
  out[30] = 0.f; out[31] = 0.f;
}

// ---------------------------------------------------------------------------
// keys pass 1: store (dd - diag) per (n,h,m); global max of dd per head
// ---------------------------------------------------------------------------
__global__ void kd_pass1(const float* __restrict__ Kmat, const float* __restrict__ proj,
                         float* __restrict__ kp_tmp, int* __restrict__ kstab_enc) {
  __shared__ float sproj[Mm * Dd];
  __shared__ int smax[Hh];
  int t = threadIdx.x;
  for (int i = t; i < Mm * Dd; i += blockDim.x) sproj[i] = proj[i];
  if (t < Hh) smax[t] = INT_MIN;
  __syncthreads();
  int unit = blockIdx.x * blockDim.x + t;
  if (unit < Nn * Hh) {
    int n = unit >> 2, h = unit & 3;
    const float* krow = Kmat + (size_t)n * HD + h * Dd;
    float data[Dd];
    float diag = 0.f;
    for (int d = 0; d < Dd; ++d) { float x = krow[d] * SCALE_QK; data[d] = x; diag += x * x; }
    diag *= 0.5f;
    float locmax = -1e30f;
    float* out = kp_tmp + (size_t)unit * Mm;
#pragma unroll 1
    for (int m = 0; m < Mm; ++m) {
      const float* pr = sproj + m * Dd;
      float s = 0.f;
      for (int d = 0; d < Dd; ++d) s += data[d] * pr[d];
      locmax = fmaxf(locmax, s);
      out[m] = s - diag;
    }
    atomicMax(&smax[h], enc_f(locmax));
  }
  __syncthreads();
  if (t < Hh) atomicMax(&kstab_enc[t], smax[t]);
}

// keys pass 2: kp = ratio * (exp(tmp - kstab[h]) + eps)
__global__ void kp_pass2(float* __restrict__ kp, const int* __restrict__ kstab_enc) {
  int idx = blockIdx.x * blockDim.x + threadIdx.x;
  if (idx >= Nn * Hh * Mm) return;
  int h = (idx / Mm) & 3;
  float stab = dec_f(kstab_enc[h]);
  kp[idx] = RATIO * (expf(kp[idx] - stab) + EPSV);
}

// ---------------------------------------------------------------------------
// kvs[h,k,m,d] = sum_n kp[n,h,m]*g[n,h,k]*v[n,h,d] ;  ksum[h,k,m] = sum kp*g
// grid = H*K*NCHUNK, atomic accumulate into zero-initialized buffers
// layouts padded to MP=32 rows of m (rows 30,31 stay zero)
// ---------------------------------------------------------------------------
__global__ void kvs_kernel(const float* __restrict__ kp, const float* __restrict__ V,
                           const float* __restrict__ G, float* __restrict__ kvs,
                           float* __restrict__ ksum) {
  int c = blockIdx.x;
  int chunk = c % NCHUNK;
  int hk = c / NCHUNK;           // h*K + k
  int h = hk / Kk, kk = hk % Kk;
  __shared__ float skp[Mm];
  __shared__ float sv[Dd];
  __shared__ float sg;
  int t = threadIdx.x;
  float acc[8]  = {0,0,0,0,0,0,0,0};
  float acc2[8] = {0,0,0,0,0,0,0,0};
  const int CH = Nn / NCHUNK;   // 1000
  int n0 = chunk * CH;
  for (int n = n0; n < n0 + CH; ++n) {
    if (t < Mm)                skp[t]      = kp[(size_t)(n * Hh + h) * Mm + t];
    else if (t >= 32 && t < 96) sv[t - 32] = V[(size_t)n * HD + h * Dd + (t - 32)];
    else if (t == 96)           sg         = expf(G[(size_t)(n * Hh + h) * Kk + kk]);
    __syncthreads();
    float g = sg;
#pragma unroll
    for (int j = 0; j < 8; ++j) {
      int e = t + 256 * j;
      int m = e >> 6, d = e & 63;
      if (m < Mm) {
        float kg = skp[m] * g;
        acc[j] += kg * sv[d];
        if (d == 0) acc2[j] += kg;
      }
    }
    __syncthreads();
  }
#pragma unroll
  for (int j = 0; j < 8; ++j) {
    int e = t + 256 * j;
    int m = e >> 6, d = e & 63;
    atomicAdd(&kvs[(size_t)(hk * MP + m) * Dd + d], acc[j]);
    if (d == 0) atomicAdd(&ksum[hk * MP + m], acc2[j]);
  }
}

// den[n,h,k] = sum_m qp[n,h,m] * ksum[h,k,m]
__global__ void den_kernel(const float* __restrict__ qp, const float* __restrict__ ksum,
                           float* __restrict__ den) {
  __shared__ float ss[Hh * Kk * MP];
  for (int i = threadIdx.x; i < Hh * Kk * MP; i += blockDim.x) ss[i] = ksum[i];
  __syncthreads();
  int idx = blockIdx.x * blockDim.x + threadIdx.x;
  if (idx >= Nn * Hh * Kk) return;
  int kk = idx % Kk;
  int h  = (idx / Kk) & 3;
  int n  = idx / (Hh * Kk);
  const float* qpp = qp + (size_t)(n * Hh + h) * MP;
  const float* sp  = ss + (h * Kk + kk) * MP;
  float s = 0.f;
  for (int m = 0; m < Mm; ++m) s += qpp[m] * sp[m];
  den[idx] = s;
}

// ---------------------------------------------------------------------------
// attention combine via WMMA: for each (n-tile, h, d-tile), loop k:
//   num = qp[16x32] @ kvs[h,k][32x16]; out += num / den[n,h,k];  out *= 1/K
// ---------------------------------------------------------------------------
__global__ void attn_kernel(const float* __restrict__ qp, const float* __restrict__ kvs,
                            const float* __restrict__ den, float* __restrict__ attn) {
  int wave = threadIdx.x >> 5;
  int rowTile = blockIdx.x * 4 + wave;
  if (rowTile >= NTILES) return;
  int h = blockIdx.y;
  int dBase = blockIdx.z * 16;
  int lane = threadIdx.x & 31;
  int half = lane >> 4, lidx = lane & 15;
  int rowBase = rowTile * 16;

  v2f a[8];
#pragma unroll
  for (int s = 0; s < 8; ++s) {
    const float* p = qp + (size_t)((rowBase + lidx) * Hh + h) * MP + s * 4 + half * 2;
    a[s].x = p[0]; a[s].y = p[1];
  }
  v8f out = {};
  for (int kk = 0; kk < Kk; ++kk) {
    v8f cf = {};
    const float* kb = kvs + (size_t)((h * Kk + kk) * MP) * Dd + dBase + lidx;
#pragma unroll
    for (int s = 0; s < 8; ++s) {
      int mrow = s * 4 + half * 2;
      v2f b;
      b.x = kb[mrow * Dd];
      b.y = kb[(mrow + 1) * Dd];
      cf = wmma4(a[s], b, cf);
    }
#pragma unroll
    for (int r = 0; r < 8; ++r) {
      int n = rowBase + r + half * 8;
      float invd = 1.0f / den[(size_t)(n * Hh + h) * Kk + kk];
      out[r] += cf[r] * invd;
    }
  }
#pragma unroll
  for (int r = 0; r < 8; ++r) {
    int n = rowBase + r + half * 8;
    attn[(size_t)n * HD + h * Dd + dBase + lidx] = out[r] * (1.0f / Kk);
  }
}

// ---------------------------------------------------------------------------
// graph side: degrees, edge norm, scatter conv
// ---------------------------------------------------------------------------
__global__ void degree_kernel(const int* __restrict__ ei, float* __restrict__ din,
                              float* __restrict__ dout) {
  int e = blockIdx.x * blockDim.x + threadIdx.x;
  if (e >= Ee) return;
  atomicAdd(&dout[ei[e]], 1.0f);        // row
  atomicAdd(&din[ei[Ee + e]], 1.0f);    // col
}

__global__ void dn_kernel(const int* __restrict__ ei, const float* __restrict__ din,
                          const float* __restrict__ dout, float* __restrict__ dn) {
  int e = blockIdx.x * blockDim.x + threadIdx.x;
  if (e >= Ee) return;
  int r = ei[e], c = ei[Ee + e];
  dn[e] = sqrtf(1.0f / din[c]) * sqrtf(1.0f / dout[r]);
}

__global__ void conv_kernel(const int* __restrict__ ei, const float* __restrict__ V,
                            const float* __restrict__ dn, float* __restrict__ conv) {
  int e = blockIdx.x;
  int c = threadIdx.x;                   // channel 0..255
  int r  = ei[e];
  int cl = ei[Ee + e];
  float d = dn[e];
  atomicAdd(&conv[(size_t)cl * HD + c], V[(size_t)r * HD + c] * d);
}

// ---------------------------------------------------------------------------
// output GEMM, fusing z_next = attn + sigmoid(b[h]) * conv on the A-side
// out[N,64] = z_next[N,256] @ Wo[64,256]^T + Wob
// ---------------------------------------------------------------------------
__global__ void out_gemm(const float* __restrict__ attn, const float* __restrict__ conv,
                         const float* __restrict__ bvec, const float* __restrict__ Wo,
                         const float* __restrict__ Wob, float* __restrict__ out) {
  int wave = threadIdx.x >> 5;
  int rowTile = blockIdx.x * 4 + wave;
  if (rowTile >= NTILES) return;
  int colBase = blockIdx.y * 16;
  int lane = threadIdx.x & 31;
  int half = lane >> 4, lidx = lane & 15;
  int rowBase = rowTile * 16;
  float sig[Hh];
#pragma unroll
  for (int h = 0; h < Hh; ++h) sig[h] = 1.0f / (1.0f + expf(-bvec[h]));

  v8f acc = {};
  for (int kk = 0; kk < HD; kk += 4) {
    int ka = kk + half * 2;
    int hh = ka >> 6;                    // head of this k-slice (2-aligned, never crosses 64)
    size_t arow = (size_t)(rowBase + lidx) * HD + ka;
    const float* wp = Wo + (size_t)(colBase + lidx) * HD + ka;
    v2f a, b;
    a.x = attn[arow]     + conv[arow]     * sig[hh];
    a.y = attn[arow + 1] + conv[arow + 1] * sig[hh];
    b.x = wp[0]; b.y = wp[1];
    acc = wmma4(a, b, acc);
  }
  float bo = Wob[colBase + lidx];
#pragma unroll
  for (int r = 0; r < 8; ++r) {
    int n = rowBase + r + half * 8;
    out[(size_t)n * Dd + colBase + lidx] = acc[r] + bo;
  }
}

__global__ void zero_f32(float* __restrict__ p, int count) {
  int i = blockIdx.x * blockDim.x + threadIdx.x;
  if (i < count) p[i] = 0.f;
}
__global__ void init_kstab(int* __restrict__ p) {
  if (threadIdx.x < Hh) p[threadIdx.x] = INT_MIN;
}

// ---------------------------------------------------------------------------
// Workspace layout (float offsets)
// ---------------------------------------------------------------------------
constexpr size_t OFF_Q     = 0;                          // [N,256]  -> reused as attn
constexpr size_t OFF_K     = OFF_Q  + (size_t)Nn * HD;   // [N,256]  -> reused as conv
constexpr size_t OFF_V     = OFF_K  + (size_t)Nn * HD;   // [N,256]
constexpr size_t OFF_QP    = OFF_V  + (size_t)Nn * HD;   // [N,H,32]
constexpr size_t OFF_KP    = OFF_QP + (size_t)Nn * Hh * MP;   // [N,H,30]
constexpr size_t OFF_KVS   = OFF_KP + (size_t)Nn * Hh * Mm;   // [H,K,32,64]
constexpr size_t OFF_KSUM  = OFF_KVS + (size_t)Hh * Kk * MP * Dd; // [H,K,32]
constexpr size_t OFF_DEN   = OFF_KSUM + (size_t)Hh * Kk * MP;     // [N,H,K]
constexpr size_t OFF_DN    = OFF_DEN + (size_t)Nn * Hh * Kk;      // [E]
constexpr size_t OFF_DIN   = OFF_DN  + (size_t)Ee;                // [N]
constexpr size_t OFF_DOUT  = OFF_DIN + (size_t)Nn;                // [N]
constexpr size_t OFF_KSTAB = OFF_DOUT + (size_t)Nn;               // [4] (int)

extern "C" void kernel_launch(void* const* d_in, const int* in_sizes, int n_in,
                              void* d_out, int out_size, void* d_ws, size_t ws_size,
                              hipStream_t stream) {
  const float* z    = (const float*)d_in[0];
  const int*   ei   = (const int*)  d_in[1];
  const float* Wq_w = (const float*)d_in[2];
  const float* Wq_b = (const float*)d_in[3];
  const float* Wk_w = (const float*)d_in[4];
  const float* Wk_b = (const float*)d_in[5];
  const float* Wv_w = (const float*)d_in[6];
  const float* Wv_b = (const float*)d_in[7];
  const float* Wo_w = (const float*)d_in[8];
  const float* Wo_b = (const float*)d_in[9];
  const float* bvec = (const float*)d_in[10];
  const float* proj = (const float*)d_in[11];
  const float* gum  = (const float*)d_in[12];

  float* ws = (float*)d_ws;
  float* q    = ws + OFF_Q;
  float* k    = ws + OFF_K;
  float* v    = ws + OFF_V;
  float* qp   = ws + OFF_QP;
  float* kp   = ws + OFF_KP;
  float* kvs  = ws + OFF_KVS;
  float* ksum = ws + OFF_KSUM;
  float* den  = ws + OFF_DEN;
  float* dn   = ws + OFF_DN;
  float* din  = ws + OFF_DIN;
  float* dou  = ws + OFF_DOUT;
  int*   kstab = (int*)(ws + OFF_KSTAB);
  float* attn = q;   // q consumed after qp_kernel
  float* conv = k;   // k consumed after kd_pass1
  float* out  = (float*)d_out;

  dim3 gTile((NTILES + 3) / 4);

  // 1) QKV projections (WMMA)
  gemm_xwt<<<dim3(gTile.x, HD / 16), 128, 0, stream>>>(z, Wq_w, Wq_b, q, CIN, HD);
  gemm_xwt<<<dim3(gTile.x, HD / 16), 128, 0, stream>>>(z, Wk_w, Wk_b, k, CIN, HD);
  gemm_xwt<<<dim3(gTile.x, HD / 16), 128, 0, stream>>>(z, Wv_w, Wv_b, v, CIN, HD);

  // 2) query features
  qp_kernel<<<(Nn * Hh + 255) / 256, 256, 0, stream>>>(q, proj, qp);

  // 3) key features (global stabilizer)
  init_kstab<<<1, 32, 0, stream>>>(kstab);
  kd_pass1<<<(Nn * Hh + 255) / 256, 256, 0, stream>>>(k, proj, kp, kstab);
  kp_pass2<<<(Nn * Hh * Mm + 255) / 256, 256, 0, stream>>>(kp, kstab);

  // 4) kvs / ksum reduction over nodes
  zero_f32<<<((Hh * Kk * MP * Dd + Hh * Kk * MP) + 255) / 256, 256, 0, stream>>>(
      kvs, Hh * Kk * MP * Dd + Hh * Kk * MP);  // kvs and ksum are contiguous
  kvs_kernel<<<Hh * Kk * NCHUNK, 256, 0, stream>>>(kp, v, gum, kvs, ksum);

  // 5) denominators, then WMMA attention combine
  den_kernel<<<(Nn * Hh * Kk + 255) / 256, 256, 0, stream>>>(qp, ksum, den);
  attn_kernel<<<dim3(gTile.x, Hh, Dd / 16), 128, 0, stream>>>(qp, kvs, den, attn);

  // 6) graph conv branch
  zero_f32<<<(2 * Nn + 255) / 256, 256, 0, stream>>>(din, 2 * Nn);  // din+dout contiguous
  degree_kernel<<<(Ee + 255) / 256, 256, 0, stream>>>(ei, din, dou);
  dn_kernel<<<(Ee + 255) / 256, 256, 0, stream>>>(ei, din, dou, dn);
  zero_f32<<<((size_t)Nn * HD + 255) / 256, 256, 0, stream>>>(conv, Nn * HD);
  conv_kernel<<<Ee, HD, 0, stream>>>(ei, v, dn, conv);

  // 7) fused output projection (WMMA)
  out_gemm<<<dim3(gTile.x, Dd / 16), 128, 0, stream>>>(attn, conv, bvec, Wo_w, Wo_b, out);
}